// Wiener_3D_67997922230652
// MI455X (gfx1250) — compile-verified
//
#include <hip/hip_runtime.h>
#include <math.h>

#define BXP     32
#define PSTRIDE 8
#define NP      69          // (576-32)/8+1 patches per dim
#define IMG_H   512
#define IMG_W   512
#define NB      4
#define NC      3
#define LDP     33          // LDS row stride (pad to dodge bank conflicts)
#define W_FFT_F   0.3f
#define W_INT_F   0.3f
#define EPSF      1e-15f

typedef __attribute__((ext_vector_type(2))) float v2f;
typedef __attribute__((ext_vector_type(8))) float v8f;

__device__ __forceinline__ int reflect_idx(int o, int n) {
  if (o < 0)  o = -o;            // 'reflect' (no edge repeat); pad(32) < n so one bounce
  if (o >= n) o = 2 * n - 2 - o;
  return o;
}

// Preload A-operand fragments of M (rows rm..rm+15, K=0..31) into registers.
__device__ __forceinline__ void load_aslab(const float* M, v2f f[8], int rm) {
  const int lane = threadIdx.x & 31, hi = lane >> 4, lo = lane & 15;
#pragma unroll
  for (int kk = 0; kk < 8; ++kk) {
    const int ka = kk * 4 + 2 * hi;
    f[kk][0] = M[(rm + lo) * LDP + ka];
    f[kk][1] = M[(rm + lo) * LDP + ka + 1];
  }
}
// Preload B-operand fragments of M (K=0..31, cols cn..cn+15) into registers.
__device__ __forceinline__ void load_bslab(const float* M, v2f f[8], int cn) {
  const int lane = threadIdx.x & 31, hi = lane >> 4, lo = lane & 15;
#pragma unroll
  for (int kk = 0; kk < 8; ++kk) {
    const int ka = kk * 4 + 2 * hi;
    f[kk][0] = M[ka * LDP + cn + lo];
    f[kk][1] = M[(ka + 1) * LDP + cn + lo];
  }
}

// acc += (cached A-slab) x (sB * B[0..31][cn..cn+15]);  B in LDS, stride LDP.
__device__ __forceinline__ void mm_cachedA(const v2f a[8], const float* B,
                                           v8f& acc, int cn, float sB) {
  const int lane = threadIdx.x & 31, hi = lane >> 4, lo = lane & 15;
#pragma unroll
  for (int kk = 0; kk < 8; ++kk) {
    const int ka = kk * 4 + 2 * hi;
    v2f b;
    b[0] = sB * B[ka * LDP + cn + lo];
    b[1] = sB * B[(ka + 1) * LDP + cn + lo];
    acc = __builtin_amdgcn_wmma_f32_16x16x4_f32(false, a[kk], false, b,
                                                (short)0, acc, false, false);
  }
}
// acc += (sA * A[rm..rm+15][0..31]) x (cached B-slab);  A in LDS, stride LDP.
__device__ __forceinline__ void mm_cachedB(const float* A, const v2f bfr[8],
                                           v8f& acc, int rm, float sA) {
  const int lane = threadIdx.x & 31, hi = lane >> 4, lo = lane & 15;
#pragma unroll
  for (int kk = 0; kk < 8; ++kk) {
    const int ka = kk * 4 + 2 * hi;
    v2f a;
    a[0] = sA * A[(rm + lo) * LDP + ka];
    a[1] = sA * A[(rm + lo) * LDP + ka + 1];
    acc = __builtin_amdgcn_wmma_f32_16x16x4_f32(false, a, false, bfr[kk],
                                                (short)0, acc, false, false);
  }
}

__device__ __forceinline__ void store_tile(float* D, const v8f& acc, int rm, int cn) {
  const int lane = threadIdx.x & 31, hi = lane >> 4, lo = lane & 15;
#pragma unroll
  for (int v = 0; v < 8; ++v)
    D[(rm + v + 8 * hi) * LDP + cn + lo] = acc[v];
}

__global__ __launch_bounds__(128)
void wiener_patch_kernel(const float* __restrict__ I, const float* __restrict__ S,
                         float* __restrict__ num, float* __restrict__ mask) {
  __shared__ float Cre[32 * LDP], Cim[32 * LDP];           // DFT matrix C = Cre + i*Cim
  __shared__ float Xw[NC][32 * LDP];                       // windowed patches
  __shared__ float Yr[32 * LDP], Yi[32 * LDP];             // stage temporaries
  __shared__ float Fr[NC][32 * LDP], Fi[NC][32 * LDP];     // spectra / filtered spectra
  __shared__ float w1f[32], w1i[32];
  __shared__ float red[4], red2[4];
  __shared__ float mean_p[NC], stdm[NC], mw2s;

  const int tid = threadIdx.x;
  const int wv  = tid >> 5;                 // wave id 0..3  -> fixed 16x16 tile
  const int rm  = (wv >> 1) * 16;           // tile row base
  const int cn  = (wv & 1) * 16;            // tile col base
  const int blk = blockIdx.x;
  const int px = blk % NP;
  const int py = (blk / NP) % NP;
  const int b  = blk / (NP * NP);

  // 1D Gaussian windows (W_FFT == W_INTERP == 0.3, but keep both generic)
  if (tid < 32) {
    float t = (float)tid - 15.5f;
    w1f[tid] = __expf(-t * t / (W_FFT_F * 256.0f));
    w1i[tid] = __expf(-t * t / (W_INT_F * 256.0f));
  }
  // DFT matrix: C[j,k] = exp(-2*pi*i*j*k/32)
  for (int e = tid; e < 1024; e += 128) {
    int j = e >> 5, k = e & 31;
    float ang = (float)(j * k) * (6.28318530717958647692f / 32.0f);
    float s, c;
    __sincosf(ang, &s, &c);
    Cre[j * LDP + k] = c;
    Cim[j * LDP + k] = -s;
  }
  __syncthreads();
  if (tid == 0) {
    float s2 = 0.f;
    for (int i = 0; i < 32; ++i) s2 += w1f[i] * w1f[i];
    float mv = s2 * (1.0f / 32.0f);
    mw2s = mv * mv;                          // mean(win^2)
  }

  // Cache the per-wave DFT-matrix fragments in registers (invariant all kernel).
  v2f aCre[8], aCim[8], bCre[8], bCim[8];
  load_aslab(Cre, aCre, rm);
  load_aslab(Cim, aCim, rm);
  load_bslab(Cre, bCre, cn);
  load_bslab(Cim, bCim, cn);

  // Load patches (reflect pad via index math) + per-channel means of I and std
  for (int c = 0; c < NC; ++c) {
    float pI = 0.f, pS = 0.f;
    for (int q = 0; q < 8; ++q) {
      int e = tid + 128 * q;
      int m = e >> 5, n = e & 31;
      int gy = reflect_idx(py * PSTRIDE + m - BXP, IMG_H);
      int gx = reflect_idx(px * PSTRIDE + n - BXP, IMG_W);
      size_t off = ((size_t)(b * NC + c) * IMG_H + gy) * IMG_W + gx;
      float vi = I[off], vs = S[off];
      Xw[c][m * LDP + n] = vi;
      pI += vi; pS += vs;
    }
    // wave32 butterfly reduction, then 4 partials via LDS
#pragma unroll
    for (int msk = 16; msk; msk >>= 1) {
      pI += __shfl_xor(pI, msk, 32);
      pS += __shfl_xor(pS, msk, 32);
    }
    if ((tid & 31) == 0) { red[wv] = pI; red2[wv] = pS; }
    __syncthreads();
    if (tid == 0) {
      mean_p[c] = (red[0] + red[1] + red[2] + red[3]) * (1.0f / 1024.0f);
      stdm[c]   = (red2[0] + red2[1] + red2[2] + red2[3]) * (1.0f / 1024.0f);
    }
    __syncthreads();
  }

  // windowed = (patch - mean) * win
  for (int c = 0; c < NC; ++c) {
    float mu = mean_p[c];
    for (int q = 0; q < 8; ++q) {
      int e = tid + 128 * q;
      int m = e >> 5, n = e & 31;
      Xw[c][m * LDP + n] = (Xw[c][m * LDP + n] - mu) * (w1f[m] * w1f[n]);
    }
  }
  __syncthreads();

  // ---- Forward 2D DFT per channel: F = C * X * C  (C symmetric) ----
  for (int c = 0; c < NC; ++c) {
    // Stage A: Yr = Cre*X ; Yi = Cim*X  (cached A fragments)
    {
      v8f accR = {}, accI = {};
      mm_cachedA(aCre, Xw[c], accR, cn, 1.0f);
      mm_cachedA(aCim, Xw[c], accI, cn, 1.0f);
      store_tile(Yr, accR, rm, cn);
      store_tile(Yi, accI, rm, cn);
    }
    __syncthreads();
    // Stage B: Fr = Yr*Cre - Yi*Cim ; Fi = Yr*Cim + Yi*Cre  (cached B fragments)
    {
      v8f fR = {}, fI = {};
      mm_cachedB(Yr, bCre, fR, rm, 1.0f);
      mm_cachedB(Yi, bCim, fR, rm, -1.0f);
      mm_cachedB(Yr, bCim, fI, rm, 1.0f);
      mm_cachedB(Yi, bCre, fI, rm, 1.0f);
      store_tile(Fr[c], fR, rm, cn);
      store_tile(Fi[c], fI, rm, cn);
    }
    __syncthreads();
  }

  // ---- 3-point channel DFT, Wiener gain, inverse 3-point DFT (pointwise) ----
  {
    const float s3 = 0.86602540378443864676f;  // sqrt(3)/2
    const float mw2 = mw2s;
    float pv[3];
#pragma unroll
    for (int u = 0; u < 3; ++u) { float sp = stdm[u]; pv[u] = mw2 * 1024.0f * sp * sp; }
    for (int q = 0; q < 8; ++q) {
      int e = tid + 128 * q;
      int m = e >> 5, n = e & 31;
      int o = m * LDP + n;
      float f0r = Fr[0][o], f0i = Fi[0][o];
      float f1r = Fr[1][o], f1i = Fi[1][o];
      float f2r = Fr[2][o], f2i = Fi[2][o];
      // w = exp(-2pi i/3) = (-1/2, -s3); w^2 = (-1/2, +s3)
      float gr[3], gi[3];
      gr[0] = f0r + f1r + f2r;                   gi[0] = f0i + f1i + f2i;
      gr[1] = f0r + (-0.5f*f1r + s3*f1i) + (-0.5f*f2r - s3*f2i);
      gi[1] = f0i + (-s3*f1r - 0.5f*f1i) + ( s3*f2r - 0.5f*f2i);
      gr[2] = f0r + (-0.5f*f1r - s3*f1i) + (-0.5f*f2r + s3*f2i);
      gi[2] = f0i + ( s3*f1r - 0.5f*f1i) + (-s3*f2r - 0.5f*f2i);
#pragma unroll
      for (int u = 0; u < 3; ++u) {
        float pss = gr[u]*gr[u] + gi[u]*gi[u] + EPSF;
        float hg  = fmaxf(pss - pv[u], 0.0f) / pss;
        gr[u] *= hg; gi[u] *= hg;
      }
      // inverse 3-pt DFT with 1/3: v = exp(+2pi i/3) = (-1/2, +s3); v^2 = (-1/2, -s3)
      const float th = (1.0f / 3.0f);
      float h0r = (gr[0] + gr[1] + gr[2]) * th;
      float h0i = (gi[0] + gi[1] + gi[2]) * th;
      float a1r = -0.5f*gr[1] - s3*gi[1], a1i =  s3*gr[1] - 0.5f*gi[1];  // G1*v
      float a2r = -0.5f*gr[2] + s3*gi[2], a2i = -s3*gr[2] - 0.5f*gi[2];  // G2*v^2
      float h1r = (gr[0] + a1r + a2r) * th, h1i = (gi[0] + a1i + a2i) * th;
      float b1r = -0.5f*gr[1] + s3*gi[1], b1i = -s3*gr[1] - 0.5f*gi[1];  // G1*v^2
      float b2r = -0.5f*gr[2] - s3*gi[2], b2i =  s3*gr[2] - 0.5f*gi[2];  // G2*v
      float h2r = (gr[0] + b1r + b2r) * th, h2i = (gi[0] + b1i + b2i) * th;
      Fr[0][o] = h0r; Fi[0][o] = h0i;
      Fr[1][o] = h1r; Fi[1][o] = h1i;
      Fr[2][o] = h2r; Fi[2][o] = h2i;
    }
  }
  __syncthreads();

  // ---- Inverse 2D DFT per channel: X = (1/1024) * conj(C) * H * conj(C), real part ----
  for (int c = 0; c < NC; ++c) {
    // Stage U: Ur = Cre*Hr + Cim*Hi -> Yr ; Ui = Cre*Hi - Cim*Hr -> Yi
    {
      v8f uR = {}, uI = {};
      mm_cachedA(aCre, Fr[c], uR, cn, 1.0f);
      mm_cachedA(aCim, Fi[c], uR, cn, 1.0f);
      mm_cachedA(aCre, Fi[c], uI, cn, 1.0f);
      mm_cachedA(aCim, Fr[c], uI, cn, -1.0f);
      store_tile(Yr, uR, rm, cn);
      store_tile(Yi, uI, rm, cn);
    }
    __syncthreads();
    // Stage X: real(X) = Ur*Cre + Ui*Cim ; fused epilogue with overlap-add atomics
    {
      v8f acc = {};
      mm_cachedB(Yr, bCre, acc, rm, 1.0f);
      mm_cachedB(Yi, bCim, acc, rm, 1.0f);
      const int lane = threadIdx.x & 31;
      const int hi = lane >> 4, lo = lane & 15;
      float mu = mean_p[c];
#pragma unroll
      for (int v = 0; v < 8; ++v) {
        int m = rm + v + 8 * hi, n = cn + lo;
        float winmn = w1f[m] * w1f[n];
        float winim = w1i[m] * w1i[n];
        float val  = acc[v] * (1.0f / 1024.0f);
        float filt = (val + mu * winmn) * winim;
        int gy = py * PSTRIDE + m - BXP;
        int gx = px * PSTRIDE + n - BXP;
        if ((unsigned)gy < (unsigned)IMG_H && (unsigned)gx < (unsigned)IMG_W) {
          atomicAdd(&num[((size_t)(b * NC + c) * IMG_H + gy) * IMG_W + gx], filt);
          if (b == 0 && c == 0)
            atomicAdd(&mask[(size_t)gy * IMG_W + gx], winmn * winim);
        }
      }
    }
    __syncthreads();
  }
}

__global__ void zero_kernel(float* __restrict__ num, float* __restrict__ mask,
                            int n_num, int n_mask) {
  int i = blockIdx.x * blockDim.x + threadIdx.x;
  if (i < n_num)  num[i] = 0.0f;
  if (i < n_mask) mask[i] = 0.0f;
}

__global__ void finalize_kernel(float* __restrict__ num, const float* __restrict__ mask,
                                int total) {
  int i = blockIdx.x * blockDim.x + threadIdx.x;
  if (i < total) {
    int pix = i % (IMG_H * IMG_W);
    num[i] = (num[i] + EPSF) / (mask[pix] + EPSF);
  }
}

extern "C" void kernel_launch(void* const* d_in, const int* in_sizes, int n_in,
                              void* d_out, int out_size, void* d_ws, size_t ws_size,
                              hipStream_t stream) {
  const float* I = (const float*)d_in[0];       // (4,3,512,512) f32
  const float* S = (const float*)d_in[1];       // (4,3,512,512) f32
  float* num  = (float*)d_out;                  // accumulator, then final output
  float* mask = (float*)d_ws;                   // 512*512 f32 shared mask
  const int n_num  = out_size;                  // 4*3*512*512
  const int n_mask = IMG_H * IMG_W;
  const int nz = (n_num > n_mask) ? n_num : n_mask;

  zero_kernel<<<(nz + 255) / 256, 256, 0, stream>>>(num, mask, n_num, n_mask);
  wiener_patch_kernel<<<dim3(NB * NP * NP), 128, 0, stream>>>(I, S, num, mask);
  finalize_kernel<<<(n_num + 255) / 256, 256, 0, stream>>>(num, mask, n_num);
}